// GCN_13675175871111
// MI455X (gfx1250) — compile-verified
//
#include <hip/hip_runtime.h>

typedef __attribute__((ext_vector_type(2))) float v2f;
typedef __attribute__((ext_vector_type(8))) float v8f;

#define EPS 1e-5f

// ------------------------------------------------------------------
// GEMM: C[N x M] = A[N x K] * B[K x M], K = 128, full-fp32 WMMA.
// grid.x = N/16, block = 32 * (M/16) threads (one wave per 16x16 tile).
// ------------------------------------------------------------------
template <int M>
__global__ void gemm_wmma(const float* __restrict__ A,
                          const float* __restrict__ B,
                          float* __restrict__ C, int N) {
  constexpr int K = 128;
  // padded stride: 132 floats/row -> word stride 132 % 64 = 4 banks, conflict-free
  __shared__ float As[16 * 132];

  const int lane = threadIdx.x & 31;
  const int wave = threadIdx.x >> 5;
  const int row0 = blockIdx.x << 4;
  const int col0 = wave << 4;
  const int half = lane >> 4;   // 0: lanes 0-15, 1: lanes 16-31
  const int idx  = lane & 15;   // M index for A-frag, N index for B-frag / C

  // cooperative stage of the 16 x 128 A tile
  for (int i = threadIdx.x; i < 16 * K; i += blockDim.x) {
    int r = i >> 7;
    int c = i & 127;
    As[r * 132 + c] = A[(size_t)(row0 + r) * K + c];
  }
  __syncthreads();

  const v2f* As2 = (const v2f*)As;  // row stride 66 v2f
  v8f acc = {};
#pragma unroll
  for (int k0 = 0; k0 < K; k0 += 4) {
    const int ka = k0 + 2 * half;               // K pair handled by this half-wave
    v2f a = As2[idx * 66 + (ka >> 1)];          // A[m=idx][ka], A[m=idx][ka+1]
    v2f b;
    b.x = B[(size_t)ka * M + col0 + idx];       // B[ka][n=idx]
    b.y = B[(size_t)(ka + 1) * M + col0 + idx]; // B[ka+1][n=idx]
    acc = __builtin_amdgcn_wmma_f32_16x16x4_f32(
        /*neg_a=*/false, a, /*neg_b=*/false, b,
        /*c_mod=*/(short)0, acc, /*reuse_a=*/false, /*reuse_b=*/false);
  }

  // C/D layout: VGPR r -> (M = r + 8*half, N = idx)
  float* Cp = C + (size_t)(row0 + half * 8) * M + col0 + idx;
#pragma unroll
  for (int r = 0; r < 8; ++r) Cp[(size_t)r * M] = acc[r];
}

// ------------------------------------------------------------------
// Degree / normalization
// ------------------------------------------------------------------
__global__ void deg_init(float* __restrict__ deg, int N) {
  int i = blockIdx.x * blockDim.x + threadIdx.x;
  if (i < N) deg[i] = 1.0f;  // self-loop
}

__global__ void deg_count(const int* __restrict__ dst, float* __restrict__ deg, int E) {
  int e = blockIdx.x * blockDim.x + threadIdx.x;
  if (e < E) atomicAdd(&deg[dst[e]], 1.0f);
}

__global__ void deg_rsqrt(float* __restrict__ dis, int N) {
  int i = blockIdx.x * blockDim.x + threadIdx.x;
  if (i < N) dis[i] = rsqrtf(dis[i]);
}

// ------------------------------------------------------------------
// Aggregation: agg = h * dis^2 + bias  (self-loop + bias), then edges
// ------------------------------------------------------------------
template <int F>
__global__ void agg_init(const float* __restrict__ h, const float* __restrict__ dis,
                         const float* __restrict__ bias, float* __restrict__ agg, int N) {
  constexpr int SH = (F == 128) ? 7 : 6;
  int i = blockIdx.x * blockDim.x + threadIdx.x;
  if (i >= N * F) return;
  int n = i >> SH;
  int f = i & (F - 1);
  float d = dis[n];
  agg[i] = h[i] * d * d + bias[f];
}

template <int F>
__global__ void agg_edges(const float* __restrict__ h, const float* __restrict__ dis,
                          const int* __restrict__ src, const int* __restrict__ dst,
                          float* __restrict__ agg, int E) {
  constexpr int VEC = F / 32;  // features per lane (4 for F=128, 2 for F=64)
  int gid  = blockIdx.x * blockDim.x + threadIdx.x;
  int e    = gid >> 5;
  int lane = gid & 31;
  if (e >= E) return;
  int s = src[e], d = dst[e];
  float c = dis[s] * dis[d];
  const float* __restrict__ hp = h + (size_t)s * F + lane * VEC;
  float* __restrict__ ap = agg + (size_t)d * F + lane * VEC;
#pragma unroll
  for (int k = 0; k < VEC; ++k) atomicAdd(ap + k, hp[k] * c);
}

// ------------------------------------------------------------------
// BatchNorm (F = 128): sum/sumsq -> scale/shift -> apply
// ------------------------------------------------------------------
__global__ void bn_zero(float* __restrict__ acc) {
  acc[threadIdx.x] = 0.0f;  // 256 floats: sum[128] + sumsq[128]
}

__global__ void bn_stats(const float* __restrict__ X, float* __restrict__ sum,
                         float* __restrict__ sumsq, int N) {
  const int F = 128;
  int f    = threadIdx.x & (F - 1);
  int rsub = threadIdx.x >> 7;         // 0..1 (blockDim = 256)
  float s = 0.0f, q = 0.0f;
  for (int n = blockIdx.x * 2 + rsub; n < N; n += gridDim.x * 2) {
    float v = X[(size_t)n * F + f];
    s += v;
    q += v * v;
  }
  __shared__ float ls[256], lq[256];
  ls[threadIdx.x] = s;
  lq[threadIdx.x] = q;
  __syncthreads();
  if (rsub == 0) {
    s += ls[128 + f];
    q += lq[128 + f];
    atomicAdd(&sum[f], s);
    atomicAdd(&sumsq[f], q);
  }
}

__global__ void bn_finalize(float* __restrict__ sum, float* __restrict__ sumsq,
                            const float* __restrict__ g, const float* __restrict__ be,
                            int N) {
  int f = threadIdx.x;  // 128 threads
  float invN  = 1.0f / (float)N;
  float mean  = sum[f] * invN;
  float var   = sumsq[f] * invN - mean * mean;  // biased variance
  float scale = g[f] * rsqrtf(var + EPS);
  sum[f]   = scale;              // reuse as scale
  sumsq[f] = be[f] - mean * scale;  // reuse as shift
}

__global__ void bn_apply(float* __restrict__ X, const float* __restrict__ scale,
                         const float* __restrict__ shift, int N) {
  const int F = 128;
  int i = blockIdx.x * blockDim.x + threadIdx.x;
  if (i >= N * F) return;
  int f = i & (F - 1);
  X[i] = X[i] * scale[f] + shift[f];
}

// ------------------------------------------------------------------
// Orchestration
// ------------------------------------------------------------------
extern "C" void kernel_launch(void* const* d_in, const int* in_sizes, int n_in,
                              void* d_out, int out_size, void* d_ws, size_t ws_size,
                              hipStream_t stream) {
  const float* x   = (const float*)d_in[0];
  const int*   ei  = (const int*)d_in[1];
  const float* W1  = (const float*)d_in[2];
  const float* b1  = (const float*)d_in[3];
  const float* g1  = (const float*)d_in[4];
  const float* be1 = (const float*)d_in[5];
  const float* W2  = (const float*)d_in[6];
  const float* b2  = (const float*)d_in[7];
  const float* g2  = (const float*)d_in[8];
  const float* be2 = (const float*)d_in[9];
  const float* W3  = (const float*)d_in[10];
  const float* b3  = (const float*)d_in[11];
  float* out = (float*)d_out;

  const int N = in_sizes[0] / 128;
  const int E = in_sizes[1] / 2;
  const int* src = ei;
  const int* dst = ei + E;

  float* ws   = (float*)d_ws;
  float* bufA = ws;                          // N x 128 (gemm out)
  float* bufB = ws + (size_t)N * 128;        // N x 128 (agg / next input)
  float* dis  = ws + (size_t)2 * N * 128;    // N
  float* bnS  = dis + N;                     // 128 (sum -> scale)
  float* bnQ  = bnS + 128;                   // 128 (sumsq -> shift)

  const int TB = 256;
  const int gN   = (N + TB - 1) / TB;
  const int gE   = (E + TB - 1) / TB;
  const int gNF  = (N * 128 + TB - 1) / TB;
  const int gNFo = (N * 64 + TB - 1) / TB;
  const int gEW  = (E * 32 + TB - 1) / TB;   // one wave per edge

  // normalization coefficients dis = (deg+1)^-1/2  (computed once)
  deg_init<<<gN, TB, 0, stream>>>(dis, N);
  deg_count<<<gE, TB, 0, stream>>>(dst, dis, E);
  deg_rsqrt<<<gN, TB, 0, stream>>>(dis, N);

  // ---- layer 1: gcn_conv(x, W1, b1) + BN(g1, be1) ----
  gemm_wmma<128><<<N / 16, 256, 0, stream>>>(x, W1, bufA, N);
  agg_init<128><<<gNF, TB, 0, stream>>>(bufA, dis, b1, bufB, N);
  agg_edges<128><<<gEW, TB, 0, stream>>>(bufA, dis, src, dst, bufB, E);
  bn_zero<<<1, 256, 0, stream>>>(bnS);  // zeroes bnS[128] + bnQ[128] (contiguous)
  bn_stats<<<512, 256, 0, stream>>>(bufB, bnS, bnQ, N);
  bn_finalize<<<1, 128, 0, stream>>>(bnS, bnQ, g1, be1, N);
  bn_apply<<<gNF, TB, 0, stream>>>(bufB, bnS, bnQ, N);

  // ---- layer 2: gcn_conv(h, W2, b2) + BN(g2, be2) ----
  gemm_wmma<128><<<N / 16, 256, 0, stream>>>(bufB, W2, bufA, N);
  agg_init<128><<<gNF, TB, 0, stream>>>(bufA, dis, b2, bufB, N);
  agg_edges<128><<<gEW, TB, 0, stream>>>(bufA, dis, src, dst, bufB, E);
  bn_zero<<<1, 256, 0, stream>>>(bnS);
  bn_stats<<<512, 256, 0, stream>>>(bufB, bnS, bnQ, N);
  bn_finalize<<<1, 128, 0, stream>>>(bnS, bnQ, g2, be2, N);
  bn_apply<<<gNF, TB, 0, stream>>>(bufB, bnS, bnQ, N);

  // ---- layer 3: gcn_conv(h, W3, b3) -> d_out ----
  gemm_wmma<64><<<N / 16, 128, 0, stream>>>(bufB, W3, bufA, N);
  agg_init<64><<<gNFo, TB, 0, stream>>>(bufA, dis, b3, out, N);
  agg_edges<64><<<gEW, TB, 0, stream>>>(bufA, dis, src, dst, out, E);
}